// Attention_21406117003362
// MI455X (gfx1250) — compile-verified
//
#include <hip/hip_runtime.h>
#include <hip/hip_bf16.h>

// ---------------------------------------------------------------------------
// MI455X (gfx1250) fused additive-attention scoring.
//   pf[b,n,h]   = features[b,n,:] . Wf[h,:]          (M=65536,K=1024,N=512)
//   pq[b,h]     = query[b,:]     . Wq[h,:]
//   u[b,n]      = sum_h v[h] * tanh(pf + pq)
//   logits[b,n] = 10 * tanh(u)
// fp32 via V_WMMA_F32_16X16X4_F32; A/B tiles staged by the Tensor Data Mover
// (tensor_load_to_lds, TENSORcnt double-buffering, D#-descriptor LDS padding).
// ---------------------------------------------------------------------------

typedef __attribute__((ext_vector_type(2))) float v2f;
typedef __attribute__((ext_vector_type(8))) float v8f;
typedef __attribute__((ext_vector_type(4))) unsigned int v4u;
typedef __attribute__((ext_vector_type(4))) int v4i;
typedef __attribute__((ext_vector_type(8))) int v8i;

namespace {
constexpr int BATCH  = 16;
constexpr int NSEQ   = 4096;
constexpr int HDIM   = 512;    // output H (GEMM N)
constexpr int KDIM   = 1024;   // 2H     (GEMM K)
constexpr int M_TILE = 32;     // rows per workgroup
constexpr int KC     = 16;     // K-chunk staged in LDS
constexpr int LDSW   = KC + 2; // padded LDS row stride (TDM pad: 16 dw + 2 dw)
constexpr int NCHUNK = KDIM / KC;
}

// --------------------------- kernel 1: pq = query @ Wq.T --------------------
__global__ __launch_bounds__(256) void proj_query_kernel(
    const float* __restrict__ query, const float* __restrict__ Wq,
    float* __restrict__ pq) {
  const int idx = blockIdx.x * 256 + threadIdx.x;   // 0 .. 16*512-1
  const int b = idx >> 9;
  const int h = idx & (HDIM - 1);
  const float* __restrict__ q = query + (size_t)b * HDIM;
  const float* __restrict__ w = Wq + (size_t)h * HDIM;
  float a0 = 0.f, a1 = 0.f, a2 = 0.f, a3 = 0.f;
  for (int f = 0; f < HDIM; f += 4) {
    a0 += q[f + 0] * w[f + 0];
    a1 += q[f + 1] * w[f + 1];
    a2 += q[f + 2] * w[f + 2];
    a3 += q[f + 3] * w[f + 3];
  }
  pq[idx] = (a0 + a1) + (a2 + a3);
}

// ---------------- Tensor Data Mover: 2-D tile load descriptor ---------------
// Loads a tile_d1 x tile_d0 (rows x contiguous f32) tile from a row-major
// tensor (row stride = stride_elems f32) into LDS at lds_byte_off, inserting
// 2 DWORDs of padding after every 16 DWORDs (pad_interval=3, pad_amount=1)
// so each row lands at a stride of LDSW=18 floats.
__device__ __forceinline__ void tdm_load_tile(
    const float* gaddr, unsigned lds_byte_off, unsigned tile_d0,
    unsigned tile_d1, unsigned tensor_d0, unsigned tensor_d1,
    unsigned stride_elems) {
  const unsigned long long ga = (unsigned long long)(uintptr_t)gaddr;
  v4u g0;
  g0.x = 1u;                                   // count=1 (valid), user mode
  g0.y = lds_byte_off;                         // lds_addr
  g0.z = (unsigned)ga;                         // global_addr[31:0]
  g0.w = (unsigned)((ga >> 32) & 0x01FFFFFFu)  // global_addr[56:32]
         | (2u << 30);                         // type = 2 ("image")
  v8i g1;
  g1[0] = (int)((2u << 16)      // data_size = 2 -> 4-byte elements
                | (1u << 20)    // pad_enable
                | (3u << 22)    // pad_interval: after 16 DWORDs
                | (1u << 25));  // pad_amount:   2 DWORDs
  g1[1] = (int)((tensor_d0 & 0xFFFFu) << 16);                      // dim0 lo
  g1[2] = (int)((tensor_d0 >> 16) | ((tensor_d1 & 0xFFFFu) << 16));
  g1[3] = (int)((tensor_d1 >> 16) | ((tile_d0 & 0xFFFFu) << 16));
  g1[4] = (int)(tile_d1 & 0xFFFFu);                                // tile_dim1
  g1[5] = (int)stride_elems;                                       // d0 stride
  g1[6] = 0;
  g1[7] = 0;
  const v4i z4 = {0, 0, 0, 0};
#if __has_include(<hip/amd_detail/amd_gfx1250_TDM.h>)
  const v8i z8 = {0, 0, 0, 0, 0, 0, 0, 0};
  __builtin_amdgcn_tensor_load_to_lds(g0, g1, z4, z4, z8, 0);
#else
  __builtin_amdgcn_tensor_load_to_lds(g0, g1, z4, z4, 0);
#endif
}

// ------------------- kernel 2: fused GEMM + tanh + v-reduction --------------
__global__ __launch_bounds__(256) void fused_attn_kernel(
    const float* __restrict__ features, const float* __restrict__ Wf,
    const float* __restrict__ pq, const float* __restrict__ vvec,
    float* __restrict__ pf_out, float* __restrict__ logits_out) {
  __shared__ __attribute__((aligned(16))) float lds_A[2][M_TILE * LDSW];
  __shared__ __attribute__((aligned(16))) float lds_B[2][HDIM * LDSW];
  __shared__ __attribute__((aligned(16))) float pq_s[HDIM];
  __shared__ __attribute__((aligned(16))) float v_s[HDIM];
  __shared__ __attribute__((aligned(16))) float u_sh[M_TILE];

  const int tid   = threadIdx.x;
  const int lane  = tid & 31;
  const int wid   = tid >> 5;     // 8 waves
  const int wm    = wid >> 2;     // 0..1 : 16-row strip within the 32-row tile
  const int wn    = wid & 3;      // 0..3 : 128-col strip of H
  const int lhalf = lane >> 4;    // half-wave select (K split per ISA layout)
  const int l16   = lane & 15;

  const size_t gmBase = (size_t)blockIdx.x * M_TILE;   // global output row base
  const int b = (int)(gmBase >> 12);                   // NSEQ=4096 rows / batch

  // Stage pq-row for this batch, v, and zero the row-sum accumulators.
  pq_s[tid]       = pq[(size_t)b * HDIM + tid];
  pq_s[tid + 256] = pq[(size_t)b * HDIM + tid + 256];
  v_s[tid]        = vvec[tid];
  v_s[tid + 256]  = vvec[tid + 256];
  if (tid < M_TILE) u_sh[tid] = 0.0f;

  v8f acc[8] = {};   // 8 x (16x16 f32 C-fragments) = 16 rows x 128 cols / wave

  const float* __restrict__ featRow = features + gmBase * KDIM;
  const unsigned ldsA_off[2] = {(unsigned)(uintptr_t)&lds_A[0][0],
                                (unsigned)(uintptr_t)&lds_A[1][0]};
  const unsigned ldsB_off[2] = {(unsigned)(uintptr_t)&lds_B[0][0],
                                (unsigned)(uintptr_t)&lds_B[1][0]};

  // Prologue: TDM-stage chunk 0 into buffer 0 (wave 0 issues for the group).
  if (wid == 0) {
    tdm_load_tile(featRow, ldsA_off[0], KC, M_TILE, KDIM, M_TILE, KDIM);
    tdm_load_tile(Wf,      ldsB_off[0], KC, HDIM,   KDIM, HDIM,   KDIM);
  }

  for (int kc = 0; kc < NCHUNK; ++kc) {
    const int buf = kc & 1;
    // Issue next chunk into the other buffer (protected by last iteration's
    // trailing barrier), then drain this chunk's two TDM ops.
    if (wid == 0) {
      if (kc + 1 < NCHUNK) {
        const int kb = (kc + 1) * KC;
        tdm_load_tile(featRow + kb, ldsA_off[buf ^ 1], KC, M_TILE, KDIM,
                      M_TILE, KDIM);
        tdm_load_tile(Wf + kb, ldsB_off[buf ^ 1], KC, HDIM, KDIM, HDIM, KDIM);
        __builtin_amdgcn_s_wait_tensorcnt((short)2);
      } else {
        __builtin_amdgcn_s_wait_tensorcnt((short)0);
      }
    }
    __syncthreads();   // chunk kc visible to all waves
    // ---- compute: 4 k-steps of 4, 8 WMMA each ----
#pragma unroll
    for (int s = 0; s < KC / 4; ++s) {
      const int k0 = 4 * s + 2 * lhalf;  // ISA f32 A/B frag: half-wave K split
      v2f a = *(const v2f*)&lds_A[buf][(wm * 16 + l16) * LDSW + k0];
#pragma unroll
      for (int t = 0; t < 8; ++t) {
        v2f bb = *(const v2f*)&lds_B[buf][(wn * 128 + t * 16 + l16) * LDSW + k0];
        acc[t] = __builtin_amdgcn_wmma_f32_16x16x4_f32(
            false, a, false, bb, (short)0, acc[t], false, false);
      }
    }
    __syncthreads();   // all reads of buf done before it is re-filled
  }

  // ---- epilogue: store pf, fuse tanh + v-weighted reduction over H ----
  float srow[8];
#pragma unroll
  for (int j = 0; j < 8; ++j) srow[j] = 0.0f;

#pragma unroll
  for (int t = 0; t < 8; ++t) {
    const int n = wn * 128 + t * 16 + l16;     // C-layout: N = lane (mod 16)
    const float pqn = pq_s[n];
    const float vn  = v_s[n];
#pragma unroll
    for (int j = 0; j < 8; ++j) {
      const float c = acc[t][j];
      const int m = wm * 16 + lhalf * 8 + j;   // C-layout: M = j (+8 upper half)
      pf_out[(gmBase + m) * HDIM + n] = c;
      srow[j] += vn * tanhf(c + pqn);
    }
  }
  // Reduce each row-partial over the 16 lanes of each half-wave.
#pragma unroll
  for (int j = 0; j < 8; ++j) {
    float sv = srow[j];
#pragma unroll
    for (int off = 1; off < 16; off <<= 1) sv += __shfl_xor(sv, off, 32);
    if (l16 == 0) {  // lanes 0 and 16 carry the two row sums
      atomicAdd(&u_sh[wm * 16 + lhalf * 8 + j], sv);  // ds_add_f32
    }
  }
  __syncthreads();
  if (tid < M_TILE) {
    logits_out[gmBase + tid] = 10.0f * tanhf(u_sh[tid]);
  }
}

// ---------------------------------------------------------------------------
extern "C" void kernel_launch(void* const* d_in, const int* in_sizes, int n_in,
                              void* d_out, int out_size, void* d_ws,
                              size_t ws_size, hipStream_t stream) {
  (void)in_sizes; (void)n_in; (void)out_size; (void)ws_size;
  const float* features = (const float*)d_in[0];
  const float* query    = (const float*)d_in[1];
  const float* Wf       = (const float*)d_in[2];
  const float* Wq       = (const float*)d_in[3];
  const float* v        = (const float*)d_in[4];

  float* pf_out     = (float*)d_out;                                   // [B,N,H]
  float* logits_out = pf_out + (size_t)BATCH * NSEQ * HDIM;            // [B,1,N]
  float* pq         = (float*)d_ws;                                    // [B,H]

  proj_query_kernel<<<(BATCH * HDIM) / 256, 256, 0, stream>>>(query, Wq, pq);
  fused_attn_kernel<<<(BATCH * NSEQ) / M_TILE, 256, 0, stream>>>(
      features, Wf, pq, v, pf_out, logits_out);
}